// YatPerformerTensorCausalAttention_64793876627788
// MI455X (gfx1250) — compile-verified
//
#include <hip/hip_runtime.h>
#include <math.h>

// ---------------- problem constants ----------------
#define Bb 2
#define Ll 2048
#define Ee 512
#define Hh 8
#define Dd 64
#define Pp 32
#define Mm 8
#define Ff 256          // R*P*M = 1*32*8
#define Tt 64           // chunk length for chunked linear attention
#define NCc 32          // L / Tt
#define BH 16           // B*H
#define NTOK (Bb*Hh*Ll) // 32768

typedef float v2f __attribute__((ext_vector_type(2)));
typedef float v8f __attribute__((ext_vector_type(8)));

// Generic strided 16x16 f32 WMMA GEMM tile: D += A(16xK) * B(Kx16)
// A element (m,k) at A[m*a_rs + k*a_cs]; B element (k,n) at B[k*b_rs + n*b_cs].
// VGPR layouts per CDNA5 ISA 7.12.2:
//   A 16x4 : lane = m + 16*(k>=2), vgpr v holds K = 2*(lane>=16)+v
//   B 4x16 : lane = n + 16*(k>=2), vgpr v holds K = 2*(lane>=16)+v
//   C/D    : row = v + 8*(lane>=16), col = lane&15
__device__ __forceinline__ v8f wmma_f32_gemm(const float* A, int a_rs, int a_cs,
                                             const float* B, int b_rs, int b_cs,
                                             int K, v8f acc) {
  const int lane = threadIdx.x & 31;
  const int mn = lane & 15;
  const int kh = (lane >> 4) << 1;
  for (int k0 = 0; k0 < K; k0 += 4) {
    const int ka = k0 + kh;
    v2f a, b;
    a.x = A[mn * a_rs + ka * a_cs];
    a.y = A[mn * a_rs + (ka + 1) * a_cs];
    b.x = B[ka * b_rs + mn * b_cs];
    b.y = B[(ka + 1) * b_rs + mn * b_cs];
    acc = __builtin_amdgcn_wmma_f32_16x16x4_f32(false, a, false, b, (short)0, acc,
                                                false, false);
  }
  return acc;
}

// 16x64 tile per wave (4 accumulators): A loaded once per k-step, reused 4x.
__device__ __forceinline__ void wmma_f32_gemm_1x4(const float* A, int a_rs, int a_cs,
                                                  const float* B, int b_rs, int b_cs,
                                                  int K, v8f acc[4]) {
  const int lane = threadIdx.x & 31;
  const int mn = lane & 15;
  const int kh = (lane >> 4) << 1;
  for (int k0 = 0; k0 < K; k0 += 4) {
    const int ka = k0 + kh;
    v2f a;
    a.x = A[mn * a_rs + ka * a_cs];
    a.y = A[mn * a_rs + (ka + 1) * a_cs];
#pragma unroll
    for (int j = 0; j < 4; ++j) {
      const float* Bj = B + j * 16 * b_cs;
      v2f b;
      b.x = Bj[ka * b_rs + mn * b_cs];
      b.y = Bj[(ka + 1) * b_rs + mn * b_cs];
      acc[j] = __builtin_amdgcn_wmma_f32_16x16x4_f32(false, a, false, b, (short)0,
                                                     acc[j], false, false);
    }
  }
}

// Async LDS fill helper: each calling thread copies 16 bytes global->LDS.
// ASYNCcnt-tracked (CDNA5 GLOBAL_LOAD_ASYNC_TO_LDS_B128); generic pointers to
// __shared__ carry the LDS byte offset in the low 32 bits (ISA 10.2 aperture).
__device__ __forceinline__ void async_copy_b128(float* lds_dst, const float* gsrc) {
  unsigned lds = (unsigned)(unsigned long long)(void*)lds_dst;
  unsigned long long ga = (unsigned long long)(const void*)gsrc;
  asm volatile("global_load_async_to_lds_b128 %0, %1, off" :: "v"(lds), "v"(ga)
               : "memory");
}
__device__ __forceinline__ void async_wait0() {
  asm volatile("s_wait_asynccnt 0x0" ::: "memory");
}

// ---------------- K1: QKV projection + scatter to (B,H,L,D) ----------------
// grid (256, 24), block 32: one wave computes a 16x64 output tile.
__global__ void k_qkv(const float* __restrict__ x, const float* __restrict__ W,
                      const float* __restrict__ bias,
                      float* __restrict__ q, float* __restrict__ k, float* __restrict__ v) {
  const int mt = blockIdx.x;           // row tile over B*L = 4096
  const int nt = blockIdx.y;           // 64-col tile over 3E = 1536
  v8f acc[4] = {};
  wmma_f32_gemm_1x4(x + (long)mt * 16 * Ee, Ee, 1,
                    W + nt * 64, 3 * Ee, 1, Ee, acc);
  const int lane = threadIdx.x & 31;
  for (int j = 0; j < 4; ++j) {
    const int col = nt * 64 + j * 16 + (lane & 15);
    const int chunk = col >> 9;        // 0=q 1=k 2=v
    const int e = col & 511;
    const int h = e >> 6, d = e & 63;
    const float bb = bias[col];
    float* dst = (chunk == 0) ? q : ((chunk == 1) ? k : v);
    for (int vi = 0; vi < 8; ++vi) {
      const int row = mt * 16 + vi + ((lane >> 4) << 3);
      const int b = row >> 11, l = row & 2047;
      dst[(((long)(b * Hh + h)) * Ll + l) * Dd + d] = acc[j][vi] + bb;
    }
  }
}

// ---------------- K2: per-token features (one wave32 per token) ----------------
// CountSketch x2 -> circular conv (== rfft*rfft->irfft over P=32) -> PRF -> fuse
__global__ void k_feat(const float* __restrict__ q, const float* __restrict__ k,
                       const float* __restrict__ omega,
                       const float* __restrict__ qn, const float* __restrict__ qw,
                       const int* __restrict__ h1, const int* __restrict__ s1,
                       const int* __restrict__ h2, const int* __restrict__ s2,
                       float* __restrict__ qf, float* __restrict__ kf) {
  const int token = blockIdx.x * 8 + (threadIdx.x >> 5);
  const int lane = threadIdx.x & 31;
  if (token >= NTOK) return;
  const int h = (token / Ll) & (Hh - 1);
  const float snode = qn[0];
  const float wt = qw[0];
  const float sqrt2s = sqrtf(fmaxf(2.0f * snode, 0.0f));
  const float wscale = sqrtf(fmaxf(wt, 0.0f)) * 0.3535533905932738f; // *1/sqrt(M)
  const int mym = lane & 7;
  const float* om = omega + (long)h * Dd * Mm + mym;

  for (int which = 0; which < 2; ++which) {
    const float* src = which ? k : q;
    float* dstf = which ? kf : qf;
    float2 z = ((const float2*)(src + (long)token * Dd))[lane];
    // L2 norm over D=64 (2 elems/lane)
    float ss = z.x * z.x + z.y * z.y;
    for (int off = 16; off > 0; off >>= 1) ss += __shfl_xor(ss, off, 32);
    const float inv = 1.0f / sqrtf(ss);
    z.x *= inv; z.y *= inv;
    // CountSketch bins (lane p owns bin p) + PRF projection (lane owns m = lane&7)
    float cs1 = 0.0f, cs2 = 0.0f, proj = 0.0f;
    for (int d = 0; d < Dd; ++d) {
      const float zd = __shfl((d & 1) ? z.y : z.x, d >> 1, 32);
      const int hh1 = h1[d], sg1 = s1[d];
      const int hh2 = h2[d], sg2 = s2[d];
      cs1 += (hh1 == lane) ? (float)sg1 * zd : 0.0f;
      cs2 += (hh2 == lane) ? (float)sg2 * zd : 0.0f;
      proj += zd * om[d * Mm];
    }
    // poly tensor sketch: circular convolution over P=32, / sqrt(P)
    float ts = 0.0f;
    for (int j = 0; j < Pp; ++j) {
      const float c1 = __shfl(cs1, j, 32);
      const float c2 = __shfl(cs2, (lane - j) & 31, 32);
      ts += c1 * c2;
    }
    ts *= 0.17677669529663687f; // 1/sqrt(32)
    // PRF feature for this lane's m
    const float arg = fminf(fmaxf(proj * sqrt2s - snode, -20.0f), 20.0f);
    const float prf = expf(arg) * wscale;
    // fuse: feat[p*M + m] = ts[p] * prf[m]; lane p stores 8 contiguous floats
    float pm[8];
    for (int m = 0; m < 8; ++m) pm[m] = __shfl(prf, m, 32);
    float* outp = dstf + (long)token * Ff + lane * 8;
    float4 o0 = make_float4(ts * pm[0], ts * pm[1], ts * pm[2], ts * pm[3]);
    float4 o1 = make_float4(ts * pm[4], ts * pm[5], ts * pm[6], ts * pm[7]);
    ((float4*)outp)[0] = o0;
    ((float4*)outp)[1] = o1;
  }
}

// ---------------- K3: per-chunk K^T V (F x D) and ksum (F) ----------------
// grid = BH*NCc = 512 blocks, 256 threads (8 waves, 8 tiles each of 16x16).
// V chunk (64x64) staged in LDS via async-to-LDS; reused by all 64 tiles.
__global__ void k_chunk_kv(const float* __restrict__ kf, const float* __restrict__ v,
                           float* __restrict__ kvp, float* __restrict__ ksp) {
  __shared__ float Vlds[Tt * Dd];      // 16 KB
  const int bh = blockIdx.x / NCc;
  const int c = blockIdx.x % NCc;
  const long base_t = (long)bh * Ll + (long)c * Tt;
  const float* Kf = kf + base_t * Ff;
  const float* Vv = v + base_t * Dd;
  const int tid = threadIdx.x;
#pragma unroll
  for (int it = 0; it < (Tt * Dd) / (4 * 256); ++it) {   // 4 iters of b128
    const int e4 = it * 256 + tid;
    async_copy_b128(&Vlds[e4 * 4], Vv + e4 * 4);
  }
  async_wait0();
  __syncthreads();

  const int wave = tid >> 5;
  const int lane = tid & 31;
  float* outp = kvp + ((long)(bh * NCc + c)) * (Ff * Dd);
  for (int i = 0; i < 8; ++i) {
    const int tile = wave * 8 + i;       // 64 tiles: 16 f-tiles x 4 d-tiles
    const int ft = tile >> 2, dt = tile & 3;
    v8f acc = {};
    // A = Kf^T (F x T): (m,k) -> Kf[k*Ff + m]; B = V (T x D) from LDS
    acc = wmma_f32_gemm(Kf + ft * 16, 1, Ff, &Vlds[dt * 16], Dd, 1, Tt, acc);
    const int colD = dt * 16 + (lane & 15);
    for (int vi = 0; vi < 8; ++vi) {
      const int rowF = ft * 16 + vi + ((lane >> 4) << 3);
      outp[rowF * Dd + colD] = acc[vi];
    }
  }
  // ksum over the chunk, one f per thread
  float s = 0.0f;
  for (int t = 0; t < Tt; ++t) s += Kf[(long)t * Ff + tid];
  ksp[(bh * NCc + c) * Ff + tid] = s;
}

// ---------------- K4: exclusive prefix over chunks (kv state + ksum state) ----
__global__ void k_prefix(float* __restrict__ kvp, float* __restrict__ ksp) {
  const int idx = blockIdx.x * blockDim.x + threadIdx.x;
  const int KV_TOT = BH * Ff * Dd;    // 262144
  if (idx < KV_TOT) {
    const int bh = idx / (Ff * Dd), rem = idx % (Ff * Dd);
    float run = 0.0f;
    for (int c = 0; c < NCc; ++c) {
      const long a = ((long)(bh * NCc + c)) * (Ff * Dd) + rem;
      const float t = kvp[a];
      kvp[a] = run;
      run += t;
    }
  } else if (idx < KV_TOT + BH * Ff) {
    const int j = idx - KV_TOT;
    const int bh = j / Ff, rem = j % Ff;
    float run = 0.0f;
    for (int c = 0; c < NCc; ++c) {
      const int a = (bh * NCc + c) * Ff + rem;
      const float t = ksp[a];
      ksp[a] = run;
      run += t;
    }
  }
}

// ---------------- K5: per-chunk attention output ----------------
// O = mask(Qf Kf^T) V + Qf KV_prev ; denom = rowsum(mask(S)) + Qf . ksum_prev
#define SLD 68
__global__ void k_attn_out(const float* __restrict__ qf, const float* __restrict__ kf,
                           const float* __restrict__ v, const float* __restrict__ kvp,
                           const float* __restrict__ ksp, float* __restrict__ att) {
  __shared__ float S[Tt * SLD];        // 17.4 KB
  __shared__ float Vlds[Tt * Dd];      // 16 KB
  __shared__ float denom[Tt];
  const int bh = blockIdx.x / NCc;
  const int c = blockIdx.x % NCc;
  const long base_t = (long)bh * Ll + (long)c * Tt;
  const float* Qf = qf + base_t * Ff;
  const float* Kf = kf + base_t * Ff;
  const float* Vv = v + base_t * Dd;
  const float* KVp = kvp + ((long)(bh * NCc + c)) * (Ff * Dd);
  const float* KSp = ksp + (bh * NCc + c) * Ff;
  const int tid = threadIdx.x;
  const int wave = tid >> 5;
  const int lane = tid & 31;

  // async-stage V chunk into LDS while S is being computed
#pragma unroll
  for (int it = 0; it < (Tt * Dd) / (4 * 256); ++it) {
    const int e4 = it * 256 + tid;
    async_copy_b128(&Vlds[e4 * 4], Vv + e4 * 4);
  }

  // stage 1: S = causal-mask(Qf Kf^T), 16 tiles (4x4), 2 per wave
  for (int i = 0; i < 2; ++i) {
    const int tile = wave * 2 + i;
    const int rt = tile >> 2, ct = tile & 3;
    v8f acc = {};
    acc = wmma_f32_gemm(Qf + (long)rt * 16 * Ff, Ff, 1,
                        Kf + (long)ct * 16 * Ff, 1, Ff, Ff, acc);
    const int gc = ct * 16 + (lane & 15);
    for (int vi = 0; vi < 8; ++vi) {
      const int gr = rt * 16 + vi + ((lane >> 4) << 3);
      S[gr * SLD + gc] = (gc <= gr) ? acc[vi] : 0.0f;   // inclusive causal mask
    }
  }
  async_wait0();
  __syncthreads();

  // stage 2: denominators
  if (tid < Tt) {
    const int t = tid;
    float rs = 0.0f;
    for (int u = 0; u < Tt; ++u) rs += S[t * SLD + u];
    float dot = 0.0f;
    for (int f = 0; f < Ff; ++f) dot += Qf[(long)t * Ff + f] * KSp[f];
    denom[t] = rs + dot + 1e-6f;
  }
  __syncthreads();

  // stage 3: O = S V + Qf KV_prev ; divide by denom; write (B,L,E) layout
  const int b = bh >> 3, h = bh & 7;
  for (int i = 0; i < 2; ++i) {
    const int tile = wave * 2 + i;
    const int rt = tile >> 2, dt = tile & 3;
    v8f acc = {};
    acc = wmma_f32_gemm(&S[rt * 16 * SLD], SLD, 1, &Vlds[dt * 16], Dd, 1, Tt, acc);
    acc = wmma_f32_gemm(Qf + (long)rt * 16 * Ff, Ff, 1, KVp + dt * 16, Dd, 1, Ff, acc);
    const int d = dt * 16 + (lane & 15);
    for (int vi = 0; vi < 8; ++vi) {
      const int lt = rt * 16 + vi + ((lane >> 4) << 3);
      const int l = c * Tt + lt;
      att[((long)(b * Ll + l)) * Ee + h * Dd + d] = acc[vi] / denom[lt];
    }
  }
}

// ---------------- K6: output projection ----------------
// grid (256, 8), block 32: one wave computes a 16x64 output tile.
__global__ void k_outproj(const float* __restrict__ att, const float* __restrict__ W,
                          const float* __restrict__ bias, float* __restrict__ out) {
  const int mt = blockIdx.x;   // over B*L
  const int nt = blockIdx.y;   // 64-col tiles over E
  v8f acc[4] = {};
  wmma_f32_gemm_1x4(att + (long)mt * 16 * Ee, Ee, 1, W + nt * 64, Ee, 1, Ee, acc);
  const int lane = threadIdx.x & 31;
  for (int j = 0; j < 4; ++j) {
    const int col = nt * 64 + j * 16 + (lane & 15);
    const float bb = bias[col];
    for (int vi = 0; vi < 8; ++vi) {
      const int row = mt * 16 + vi + ((lane >> 4) << 3);
      out[(long)row * Ee + col] = acc[j][vi] + bb;
    }
  }
}

extern "C" void kernel_launch(void* const* d_in, const int* in_sizes, int n_in,
                              void* d_out, int out_size, void* d_ws, size_t ws_size,
                              hipStream_t stream) {
  const float* x     = (const float*)d_in[0];
  const float* W_qkv = (const float*)d_in[1];
  const float* b_qkv = (const float*)d_in[2];
  const float* W_out = (const float*)d_in[3];
  const float* b_out = (const float*)d_in[4];
  const float* omega = (const float*)d_in[5];
  const float* qn    = (const float*)d_in[6];
  const float* qw    = (const float*)d_in[7];
  const int*   h1    = (const int*)d_in[8];
  const int*   s1    = (const int*)d_in[9];
  const int*   h2    = (const int*)d_in[10];
  const int*   s2    = (const int*)d_in[11];

  float* ws = (float*)d_ws;
  float* q   = ws;                                // NTOK*64
  float* k   = q + (long)NTOK * Dd;               // NTOK*64
  float* v   = k + (long)NTOK * Dd;               // NTOK*64
  float* qf  = v + (long)NTOK * Dd;               // NTOK*256
  float* kf  = qf + (long)NTOK * Ff;              // NTOK*256
  float* kvp = kf + (long)NTOK * Ff;              // BH*NCc*Ff*Dd
  float* ksp = kvp + (long)BH * NCc * Ff * Dd;    // BH*NCc*Ff
  float* att = ksp + (long)BH * NCc * Ff;         // B*L*E

  k_qkv<<<dim3(256, 24), 32, 0, stream>>>(x, W_qkv, b_qkv, q, k, v);
  k_feat<<<dim3(NTOK / 8), 256, 0, stream>>>(q, k, omega, qn, qw, h1, s1, h2, s2, qf, kf);
  k_chunk_kv<<<dim3(BH * NCc), 256, 0, stream>>>(kf, v, kvp, ksp);
  {
    const int total = BH * Ff * Dd + BH * Ff;
    k_prefix<<<dim3((total + 255) / 256), 256, 0, stream>>>(kvp, ksp);
  }
  k_attn_out<<<dim3(BH * NCc), 256, 0, stream>>>(qf, kf, v, kvp, ksp, att);
  k_outproj<<<dim3(256, Ee / 64), 32, 0, stream>>>(att, W_out, b_out, (float*)d_out);
}